// HouseholderSylvesterFlow_41592463294507
// MI455X (gfx1250) — compile-verified
//
#include <hip/hip_runtime.h>
#include <cmath>

// HouseholderSylvesterFlow — fused single kernel for MI455X (gfx1250).
// Phase A per head k: P = h_tile(16x300) @ W_k(300x2336) + b_k via
//   V_WMMA_F32_16X16X4_F32 (exact fp32 matrix path), two N-tiles per wave
//   pass -> one A-operand load feeds two independent WMMA accumulation chains.
// Phase B per wave (= per batch row): Householder Q build (rank-1 updates in
//   VGPRs), triangular matvecs with diagonal-skewed conflict-free LDS access,
//   sequential z/ln_det update.

#define L      32
#define KHEADS 8
#define NH     8
#define HID    300
#define BATCH  4096
#define NB     18688

#define TILE_M 16
#define NWAVES 16
#define THREADS (NWAVES * 32)

#define PCOLS 2336
#define NTILES (PCOLS / 16)   // 146 (even)
#define PSTR  2340   // P row stride (floats): 8*PSTR % 64 == 32 -> conflict-free WMMA stores
#define QSTR  33     // Q row stride: 33 invertible mod 64 -> conflict-free both directions
#define HSTR  308    // h row stride: 4*(13m mod 16) distinct banks

#define ROFF  0
#define RTOFF 1024
#define VOFF  2048
#define BBOFF 2304

typedef float v2f __attribute__((ext_vector_type(2)));
typedef float v8f __attribute__((ext_vector_type(8)));

__device__ __forceinline__ float softplus_f(float x) {
    // stable log(1+exp(x))
    return log1pf(__expf(-fabsf(x))) + fmaxf(x, 0.0f);
}

__device__ __forceinline__ int pcol_to_wcol(int pcol, int k) {
    if      (pcol < 1024) return k * 1024 + pcol;                  // R
    else if (pcol < 2048) return 8192  + k * 1024 + (pcol - 1024); // R_t
    else if (pcol < 2304) return 16384 + k * 256  + (pcol - 2048); // V
    else                  return 18432 + k * 32   + (pcol - 2304); // Bb
}

extern "C" __global__ __launch_bounds__(THREADS)
void sylvester_flow_kernel(const float* __restrict__ z0,
                           const float* __restrict__ hmat,
                           const float* __restrict__ W,
                           const float* __restrict__ bvec,
                           float* __restrict__ out)
{
    extern __shared__ float smem[];
    float* P  = smem;                        // TILE_M * PSTR
    float* Qs = P  + TILE_M * PSTR;          // TILE_M * (L*QSTR)
    float* hL = Qs + TILE_M * (L * QSTR);    // TILE_M * HSTR
    float* zs = hL + TILE_M * HSTR;          // TILE_M * L
    float* ys = zs + TILE_M * L;
    float* ts = ys + TILE_M * L;
    float* ss = ts + TILE_M * L;

    const int tid  = threadIdx.x;
    const int wave = tid >> 5;
    const int lane = tid & 31;
    const int hi   = lane >> 4;    // half-wave id
    const int n16  = lane & 15;    // N column / A row within tile
    const int kk2  = hi * 2;       // K sub-row pair base
    const int row0 = blockIdx.x * TILE_M;

    // ---- stage h tile into LDS ----
    for (int idx = tid; idx < TILE_M * HID; idx += THREADS) {
        int r = idx / HID, c = idx - r * HID;
        hL[r * HSTR + c] = hmat[(size_t)(row0 + r) * HID + c];
    }
    // ---- stage z (wave w owns batch row w) ----
    zs[wave * L + lane] = z0[(size_t)(row0 + wave) * L + lane];

    const int myrow = wave;
    const int prow  = myrow * PSTR;
    const int qbase = myrow * (L * QSTR);
    float ln_det = 0.0f;

    __syncthreads();

    for (int k = 0; k < KHEADS; ++k) {
        // ================= Phase A: params GEMM (f32 WMMA) =================
        // Two N-tiles per pass: shared A operand, two independent acc chains.
        for (int nt0 = wave * 2; nt0 < NTILES; nt0 += NWAVES * 2) {
            int pcol0 = nt0 * 16;
            int pcol1 = pcol0 + 16;                   // NTILES even -> valid
            int wcol0 = pcol_to_wcol(pcol0, k);
            int wcol1 = pcol_to_wcol(pcol1, k);

            const float* wp0 = W + (size_t)kk2 * NB + wcol0 + n16;
            const float* wp1 = W + (size_t)kk2 * NB + wcol1 + n16;
            v8f acc0 = {};
            v8f acc1 = {};
            #pragma unroll 5
            for (int s = 0; s < HID / 4; ++s) {
                size_t roff = (size_t)s * 4 * NB;
                v2f bv0, bv1;
                bv0.x = wp0[roff];
                bv0.y = wp0[roff + NB];
                bv1.x = wp1[roff];
                bv1.y = wp1[roff + NB];
                int hb = n16 * HSTR + 4 * s + kk2;
                v2f av;
                av.x = hL[hb];
                av.y = hL[hb + 1];
                acc0 = __builtin_amdgcn_wmma_f32_16x16x4_f32(
                    false, av, false, bv0, (short)0, acc0, false, false);
                acc1 = __builtin_amdgcn_wmma_f32_16x16x4_f32(
                    false, av, false, bv1, (short)0, acc1, false, false);
            }
            float bias0 = bvec[wcol0 + n16];
            float bias1 = bvec[wcol1 + n16];
            #pragma unroll
            for (int r = 0; r < 8; ++r) {
                P[(r + 8 * hi) * PSTR + pcol0 + n16] = acc0[r] + bias0;
                P[(r + 8 * hi) * PSTR + pcol1 + n16] = acc1[r] + bias1;
            }
        }
        __syncthreads();

        // ================= Phase B: per-row flow step =================
        // Build Q = H0 H1 ... H7 applied from the right; lane holds row `lane`.
        float q[L];
        #pragma unroll
        for (int j = 0; j < L; ++j) q[j] = (j == lane) ? 1.0f : 0.0f;

        for (int hd = 0; hd < NH; ++hd) {
            int vb = prow + VOFF + hd * L;
            float vv = 0.0f, u = 0.0f;
            #pragma unroll
            for (int j = 0; j < L; ++j) {
                float vj = P[vb + j];          // broadcast read
                vv += vj * vj;
                u  += q[j] * vj;
            }
            float cu = (2.0f / vv) * u;
            #pragma unroll
            for (int j = 0; j < L; ++j) q[j] -= cu * P[vb + j];
        }
        #pragma unroll
        for (int j = 0; j < L; ++j) Qs[qbase + lane * QSTR + j] = q[j];

        // y = Q^T z  (lane computes y[lane] = column dot)
        float y = 0.0f;
        #pragma unroll
        for (int i = 0; i < L; ++i)
            y += Qs[qbase + i * QSTR + lane] * zs[myrow * L + i];
        ys[myrow * L + lane] = y;

        // diagonal terms (lane m handles dim m)
        float rtraw = P[prow + RTOFF + lane * (L + 1)];
        float rt_d  = softplus_f(rtraw) + 0.01f;
        float rraw  = P[prow + ROFF  + lane * (L + 1)];
        float rr    = rraw * rt_d;
        float mval  = softplus_f(rr) - 1.0f;         // = diag(R_t @ R)
        float rdnew = rraw + (mval - rr) / rt_d;

        // w = triu(R_t) y + Bb   (diagonal-skewed, conflict-free)
        float wv = rt_d * y + P[prow + BBOFF + lane];
        #pragma unroll
        for (int jj = 1; jj < L; ++jj) {
            int col = (lane + jj) & (L - 1);
            float e = P[prow + RTOFF + lane * L + col] * ys[myrow * L + col];
            wv += (col > lane) ? e : 0.0f;
        }
        float t = tanhf(wv);
        ts[myrow * L + lane] = t;

        // s = triu(R) t
        float sv = rdnew * t;
        #pragma unroll
        for (int jj = 1; jj < L; ++jj) {
            int col = (lane + jj) & (L - 1);
            float e = P[prow + ROFF + lane * L + col] * ts[myrow * L + col];
            sv += (col > lane) ? e : 0.0f;
        }
        ss[myrow * L + lane] = sv;

        // z += Q s  (Q rows still in registers)
        float dz = 0.0f;
        #pragma unroll
        for (int j = 0; j < L; ++j) dz += q[j] * ss[myrow * L + j];
        zs[myrow * L + lane] += dz;

        float hp = 1.0f - t * t;
        ln_det += logf(fabsf(1.0f + hp * mval) + 1e-8f);

        __syncthreads();  // protect P before next head's GEMM
    }

    out[(size_t)(row0 + myrow) * L + lane] = zs[myrow * L + lane];
    out[(size_t)BATCH * L + (size_t)(row0 + myrow) * L + lane] = ln_det;
}

extern "C" void kernel_launch(void* const* d_in, const int* in_sizes, int n_in,
                              void* d_out, int out_size, void* d_ws, size_t ws_size,
                              hipStream_t stream) {
    (void)in_sizes; (void)n_in; (void)d_ws; (void)ws_size; (void)out_size;
    const float* z0 = (const float*)d_in[0];
    const float* h  = (const float*)d_in[1];
    const float* W  = (const float*)d_in[2];
    const float* b  = (const float*)d_in[3];
    float* out = (float*)d_out;

    size_t lds_bytes = (size_t)(TILE_M * PSTR + TILE_M * L * QSTR +
                                TILE_M * HSTR + 4 * TILE_M * L) * sizeof(float);
    hipFuncSetAttribute((const void*)sylvester_flow_kernel,
                        hipFuncAttributeMaxDynamicSharedMemorySize,
                        (int)lds_bytes);
    sylvester_flow_kernel<<<BATCH / TILE_M, THREADS, lds_bytes, stream>>>(
        z0, h, W, b, out);
}